// GENConv_41609643164450
// MI455X (gfx1250) — compile-verified
//
#include <hip/hip_runtime.h>

#define GEN_EPS 1e-7f

typedef float v2f __attribute__((ext_vector_type(2)));
typedef float v8f __attribute__((ext_vector_type(8)));

// ---------------------------------------------------------------------------
// Kernel 1: zero the workspace (mx | num | den), float4-wide.
// ---------------------------------------------------------------------------
__global__ void genconv_init_ws(float4* __restrict__ ws, int n4) {
    int t = blockIdx.x * blockDim.x + threadIdx.x;
    if (t < n4) ws[t] = make_float4(0.f, 0.f, 0.f, 0.f);
}

// ---------------------------------------------------------------------------
// Kernel 2: segment max of beta*(relu(x[src])+eps) into mx[dst].
// Values are strictly positive, so the IEEE754 bit pattern is monotone as
// unsigned int -> integer atomicMax (all arrays L2-resident: 192MB L2).
// One thread per (edge, 4-feature chunk): t in [0, E*16).
// ---------------------------------------------------------------------------
__global__ void genconv_edge_max(const float* __restrict__ x,
                                 const int* __restrict__ ei,
                                 const float* __restrict__ beta_p,
                                 unsigned int* __restrict__ mx,
                                 int E) {
    int t = blockIdx.x * blockDim.x + threadIdx.x;
    if (t >= E * 16) return;
    int e = t >> 4;
    int c = (t & 15) << 2;           // feature base 0..60
    int dst = ei[e];                 // edge_index[0] = dst
    int src = ei[E + e];             // edge_index[1] = src
    float beta = beta_p[0];
    float4 xv = *(const float4*)(x + src * 64 + c);
    float v0 = (fmaxf(xv.x, 0.f) + GEN_EPS) * beta;
    float v1 = (fmaxf(xv.y, 0.f) + GEN_EPS) * beta;
    float v2 = (fmaxf(xv.z, 0.f) + GEN_EPS) * beta;
    float v3 = (fmaxf(xv.w, 0.f) + GEN_EPS) * beta;
    unsigned int* base = mx + dst * 64 + c;
    atomicMax(base + 0, __float_as_uint(v0));
    atomicMax(base + 1, __float_as_uint(v1));
    atomicMax(base + 2, __float_as_uint(v2));
    atomicMax(base + 3, __float_as_uint(v3));
}

// ---------------------------------------------------------------------------
// Kernel 3: den[dst] += exp(beta*m - mx[dst]);  num[dst] += m * exp(...)
// (softmax weight uses beta*m, the aggregated value is m itself).
// ---------------------------------------------------------------------------
__global__ void genconv_edge_sum(const float* __restrict__ x,
                                 const int* __restrict__ ei,
                                 const float* __restrict__ beta_p,
                                 const float* __restrict__ mx,
                                 float* __restrict__ num,
                                 float* __restrict__ den,
                                 int E) {
    int t = blockIdx.x * blockDim.x + threadIdx.x;
    if (t >= E * 16) return;
    int e = t >> 4;
    int c = (t & 15) << 2;
    int dst = ei[e];
    int src = ei[E + e];
    float beta = beta_p[0];
    float4 xv = *(const float4*)(x + src * 64 + c);
    float m0 = fmaxf(xv.x, 0.f) + GEN_EPS;
    float m1 = fmaxf(xv.y, 0.f) + GEN_EPS;
    float m2 = fmaxf(xv.z, 0.f) + GEN_EPS;
    float m3 = fmaxf(xv.w, 0.f) + GEN_EPS;
    float4 mxv = *(const float4*)(mx + dst * 64 + c);
    float e0 = __expf(beta * m0 - mxv.x);
    float e1 = __expf(beta * m1 - mxv.y);
    float e2 = __expf(beta * m2 - mxv.z);
    float e3 = __expf(beta * m3 - mxv.w);
    float* nb = num + dst * 64 + c;
    float* db = den + dst * 64 + c;
    atomicAdd(db + 0, e0);
    atomicAdd(db + 1, e1);
    atomicAdd(db + 2, e2);
    atomicAdd(db + 3, e3);
    atomicAdd(nb + 0, m0 * e0);
    atomicAdd(nb + 1, m1 * e1);
    atomicAdd(nb + 2, m2 * e2);
    atomicAdd(nb + 3, m3 * e3);
}

// ---------------------------------------------------------------------------
// Kernel 4: h = num/den, written in place over num (branchless: v_rcp +
// cndmask; empty segments -> 0, matching JAX segment_sum of empty segments).
// Hoisted out of the GEMM so the WMMA loop is pure loads + matrix ops.
// ---------------------------------------------------------------------------
__global__ void genconv_normalize(float4* __restrict__ num,
                                  const float4* __restrict__ den, int n4) {
    int t = blockIdx.x * blockDim.x + threadIdx.x;
    if (t >= n4) return;
    float4 n = num[t];
    float4 d = den[t];
    float4 h;
    h.x = d.x > 0.f ? n.x * __builtin_amdgcn_rcpf(d.x) : 0.f;
    h.y = d.y > 0.f ? n.y * __builtin_amdgcn_rcpf(d.y) : 0.f;
    h.z = d.z > 0.f ? n.z * __builtin_amdgcn_rcpf(d.z) : 0.f;
    h.w = d.w > 0.f ? n.w * __builtin_amdgcn_rcpf(d.w) : 0.f;
    num[t] = h;
}

// ---------------------------------------------------------------------------
// Kernel 5: out = h @ W^T + b using V_WMMA_F32_16X16X4_F32.
// One wave (32 lanes) owns a 16-row tile of h and all 64 output columns:
// 4 f32 accumulators x 16 K-steps (K=4 each) = 64 WMMA ops per wave.
//
// f32 16x16x4 operand layout (ISA 7.12.2):
//   A (16x4): lanes 0-15 -> row M=lane, VGPR0=K0, VGPR1=K1
//             lanes 16-31 -> row M=lane-16, VGPR0=K2, VGPR1=K3
//   B (4x16): mirrored (N instead of M). B[k][n] = W[n][k] since out = h@W^T.
//   C/D 16x16 f32: VGPR r -> M = r + 8*(lane>=16), N = lane&15.
// ---------------------------------------------------------------------------
__global__ void __launch_bounds__(256)
genconv_gemm_wmma(const float* __restrict__ h,
                  const float* __restrict__ W,
                  const float* __restrict__ bias,
                  float* __restrict__ out,
                  int N, int ntiles) {
    int tile = blockIdx.x * 8 + (threadIdx.x >> 5);
    if (tile >= ntiles) return;                 // wave-uniform guard: EXEC all 1s below
    int lane = threadIdx.x & 31;
    int half = lane >> 4;                       // 0 or 1
    int l16  = lane & 15;

    int arow = tile * 16 + l16;
    if (arow >= N) arow = N - 1;                // clamp (N=50000 is 16-aligned; safety)
    const float* hrow = h + arow * 64 + 2 * half;
    const float* w0 = W + (l16 +  0) * 64 + 2 * half;
    const float* w1 = W + (l16 + 16) * 64 + 2 * half;
    const float* w2 = W + (l16 + 32) * 64 + 2 * half;
    const float* w3 = W + (l16 + 48) * 64 + 2 * half;

    v8f acc0 = {}; v8f acc1 = {}; v8f acc2 = {}; v8f acc3 = {};

#pragma unroll
    for (int k0 = 0; k0 < 64; k0 += 4) {
        v2f a  = *(const v2f*)(hrow + k0);
        v2f b0 = *(const v2f*)(w0 + k0);
        v2f b1 = *(const v2f*)(w1 + k0);
        v2f b2 = *(const v2f*)(w2 + k0);
        v2f b3 = *(const v2f*)(w3 + k0);
        acc0 = __builtin_amdgcn_wmma_f32_16x16x4_f32(false, a, false, b0, (short)0, acc0, false, false);
        acc1 = __builtin_amdgcn_wmma_f32_16x16x4_f32(false, a, false, b1, (short)0, acc1, false, false);
        acc2 = __builtin_amdgcn_wmma_f32_16x16x4_f32(false, a, false, b2, (short)0, acc2, false, false);
        acc3 = __builtin_amdgcn_wmma_f32_16x16x4_f32(false, a, false, b3, (short)0, acc3, false, false);
    }

    float bj0 = bias[l16 +  0];
    float bj1 = bias[l16 + 16];
    float bj2 = bias[l16 + 32];
    float bj3 = bias[l16 + 48];

    int rowBase = tile * 16 + 8 * half;
#pragma unroll
    for (int r = 0; r < 8; ++r) {
        int row = rowBase + r;
        if (row < N) {
            float* orow = out + row * 64 + l16;
            orow[ 0] = acc0[r] + bj0;
            orow[16] = acc1[r] + bj1;
            orow[32] = acc2[r] + bj2;
            orow[48] = acc3[r] + bj3;
        }
    }
}

// ---------------------------------------------------------------------------
extern "C" void kernel_launch(void* const* d_in, const int* in_sizes, int n_in,
                              void* d_out, int out_size, void* d_ws, size_t ws_size,
                              hipStream_t stream) {
    const float* x    = (const float*)d_in[0];   // [N,64]
    const float* W    = (const float*)d_in[1];   // [64,64]
    const float* b    = (const float*)d_in[2];   // [64]
    const float* beta = (const float*)d_in[3];   // [1]
    const int*   ei   = (const int*)d_in[4];     // [2,E] row0=dst, row1=src

    float* out = (float*)d_out;
    int N = in_sizes[0] / 64;
    int E = in_sizes[4] / 2;

    float* mx  = (float*)d_ws;                   // [N,64] (positive-float bit max)
    float* num = mx  + (size_t)N * 64;           // [N,64] -> becomes h in place
    float* den = num + (size_t)N * 64;           // [N,64]

    // 1) zero workspace
    int n4all = 3 * N * 16;                      // 3*N*64 floats as float4
    genconv_init_ws<<<(n4all + 255) / 256, 256, 0, stream>>>((float4*)d_ws, n4all);

    // 2) segment max (atomic uint max on positive-float bit patterns)
    int et = E * 16;
    genconv_edge_max<<<(et + 255) / 256, 256, 0, stream>>>(
        x, ei, beta, (unsigned int*)mx, E);

    // 3) exp / weighted-exp accumulation
    genconv_edge_sum<<<(et + 255) / 256, 256, 0, stream>>>(
        x, ei, beta, mx, num, den, E);

    // 4) h = num/den in place (branchless rcp)
    int n4 = N * 16;
    genconv_normalize<<<(n4 + 255) / 256, 256, 0, stream>>>(
        (float4*)num, (const float4*)den, n4);

    // 5) out = h @ W^T + b via f32 WMMA
    int ntiles = (N + 15) / 16;
    int gblocks = (ntiles + 7) / 8;              // 8 waves per 256-thread block
    genconv_gemm_wmma<<<gblocks, 256, 0, stream>>>(num, W, b, out, N, ntiles);
}